// DiscriminativeLoss_86242943304305
// MI455X (gfx1250) — compile-verified
//
#include <hip/hip_runtime.h>

// Sum-reduction of prediction[16,8,512,512] (the one-hot masks sum to 1, so the
// einsum == prediction.sum()). target is mathematically irrelevant -> skipped,
// halving HBM traffic vs. a naive faithful implementation.
//
// Pass 1: 2048 blocks x 256 threads (8 wave32/block). Each lane streams
//         float4s (global_load_b128); a V_WMMA_F32_16X16X4_F32 with A=ones
//         accumulates the whole wave's data into an f32 16x16 accumulator.
// Pass 2: one block deterministically folds the 2048 partials.

typedef float v2f __attribute__((ext_vector_type(2)));
typedef float v8f __attribute__((ext_vector_type(8)));

#define NB 2048
#define NT 256

__global__ __launch_bounds__(NT) void DL_sum_partial(const float* __restrict__ in,
                                                     long long n,
                                                     float* __restrict__ partial) {
    const long long tid = (long long)blockIdx.x * NT + threadIdx.x;
    const long long T   = (long long)NB * NT;            // total threads
    const long long n4  = n >> 2;                        // float4 count
    const long long nfull = n4 / T;                      // uniform trip count
    const long long rem4  = n4 - nfull * T;

    const float4* __restrict__ in4 = (const float4*)in;

    v8f c = {};                 // 16x16 f32 accumulator (8 VGPRs)
    v2f a; a[0] = 1.0f; a[1] = 1.0f;   // A = ones(16x4), layout-invariant

    long long idx = tid;
    #pragma unroll 4
    for (long long i = 0; i < nfull; ++i, idx += T) {
        float4 v = in4[idx];                             // global_load_b128
        v2f b; b[0] = v.x + v.z; b[1] = v.y + v.w;       // fold 4 -> 2 per lane
        // D[m][n] = sum_k B[k][n] + C[m][n]  (A == 1 everywhere)
        c = __builtin_amdgcn_wmma_f32_16x16x4_f32(
                /*neg_a=*/false, a, /*neg_b=*/false, b,
                /*c_mod=*/(short)0, c, /*reuse_a=*/false, /*reuse_b=*/false);
    }

    // Divergent tails (EXEC restored before any further wmma use of c).
    float tacc = 0.0f;
    if (tid < rem4) {
        float4 v = in4[nfull * T + tid];
        tacc = (v.x + v.y) + (v.z + v.w);
    }
    if (tid == 0) {                                      // n % 4 leftovers
        for (long long j = n4 << 2; j < n; ++j) tacc += in[j];
    }

    // c[0]: lanes 0-15 = row0 column sums, lanes 16-31 = row8 (identical).
    // Sum over 32 lanes of 0.5*c[0] == this wave's full data total.
    float lane_val = 0.5f * c[0] + tacc;
    #pragma unroll
    for (int off = 16; off > 0; off >>= 1)
        lane_val += __shfl_xor(lane_val, off, 32);

    __shared__ float wsum[NT / 32];
    const int lane = threadIdx.x & 31;
    const int wave = threadIdx.x >> 5;
    if (lane == 0) wsum[wave] = lane_val;
    __syncthreads();
    if (threadIdx.x == 0) {
        float s = 0.0f;
        #pragma unroll
        for (int w = 0; w < NT / 32; ++w) s += wsum[w];   // fixed order
        partial[blockIdx.x] = s;
    }
}

__global__ __launch_bounds__(NT) void DL_sum_final(const float* __restrict__ partial,
                                                   float* __restrict__ out) {
    __shared__ float sm[NT];
    float s = 0.0f;
    #pragma unroll
    for (int i = threadIdx.x; i < NB; i += NT) s += partial[i];  // fixed order
    sm[threadIdx.x] = s;
    __syncthreads();
    #pragma unroll
    for (int stride = NT / 2; stride > 0; stride >>= 1) {        // fixed tree
        if (threadIdx.x < stride) sm[threadIdx.x] += sm[threadIdx.x + stride];
        __syncthreads();
    }
    if (threadIdx.x == 0) out[0] = sm[0];
}

extern "C" void kernel_launch(void* const* d_in, const int* in_sizes, int n_in,
                              void* d_out, int out_size, void* d_ws, size_t ws_size,
                              hipStream_t stream) {
    const float* pred = (const float*)d_in[0];   // prediction [16,8,512,512] f32
    // d_in[1] (target) is provably irrelevant: sum_k one_hot == 1.
    float* partial = (float*)d_ws;               // NB * 4 bytes scratch
    float* out     = (float*)d_out;              // scalar f32
    const long long n = (long long)in_sizes[0];

    DL_sum_partial<<<NB, NT, 0, stream>>>(pred, n, partial);
    DL_sum_final<<<1, NT, 0, stream>>>(partial, out);
}